// M_H_82111184765104
// MI455X (gfx1250) — compile-verified
//
#include <hip/hip_runtime.h>

// FNO spectral mixing on MI455X (gfx1250), fp32 WMMA path.
// out[b,i,:] = ifft( W_i @ fft(fh[b,i,:]) ), one workgroup per mode i.

typedef __attribute__((ext_vector_type(2))) float v2f;
typedef __attribute__((ext_vector_type(8))) float v8f;

#define BATCH  64
#define NDIM   8192
#define KDIM   64
#define NMODES 2048
#define LDSS   68   // padded LDS row stride (floats): bank = (4*row + k) % 64, conflict-free

__device__ __forceinline__ v8f wmma4(v2f a, v2f b, v8f c) {
  // V_WMMA_F32_16X16X4_F32 : D = A(16x4) * B(4x16) + C(16x16)
  return __builtin_amdgcn_wmma_f32_16x16x4_f32(
      /*neg_a=*/false, a, /*neg_b=*/false, b,
      /*c_mod=*/(short)0, c, /*reuse_a=*/false, /*reuse_b=*/false);
}

// A-fragment (16x4) and B-fragment (4x16) have mirrored layouts; both read as a
// ds_load_b64 of 2 consecutive K values from a row-major [row][k] LDS matrix
// (for B the matrix stored is B^T, i.e. [n][k]).
__device__ __forceinline__ v2f ldfrag(const float* base, int tile16, int lane, int k0) {
  const int m  = lane & 15;
  const int hi = lane >> 4;
  return *(const v2f*)(base + (tile16 + m) * LDSS + k0 + 2 * hi);
}

// C/D tile (16x16 f32): VGPR v, lane l -> (M = v + 8*(l>>4), N = l&15)
__device__ __forceinline__ void st_tile(float* base, int rt, int ct, int lane, v8f c) {
  const int n  = lane & 15;
  const int hi = lane >> 4;
#pragma unroll
  for (int v = 0; v < 8; ++v)
    base[(rt + v + 8 * hi) * LDSS + ct + n] = c[v];
}

__global__ __launch_bounds__(256) void fno_mode_kernel(
    const float* __restrict__ fh,    // [64, 8192, 64] f32
    const float* __restrict__ wts,   // [2048, 64, 64] complex64 (interleaved re,im)
    float* __restrict__ out)         // [64, 2048, 64] complex64 (interleaved)
{
  __shared__ float Cs[64 * LDSS];   // cos(2*pi*k*k'/64)
  __shared__ float Sn[64 * LDSS];   // -sin(2*pi*k*k'/64)
  __shared__ float A [64 * LDSS];   // fh[:, mode, :]   (64b x 64k)
  __shared__ float Wr[64 * LDSS];   // Re W_mode [j][l]
  __shared__ float Wi[64 * LDSS];   // Im W_mode [j][l]
  __shared__ float Gr[64 * LDSS];   // Re fft(A)
  __shared__ float Gi[64 * LDSS];   // Im fft(A)
  __shared__ float Tr[64 * LDSS];   // Re (G @ W^T)
  __shared__ float Ti[64 * LDSS];   // Im (G @ W^T)

  const int mode = blockIdx.x;
  const int tid  = threadIdx.x;
  const int wv   = tid >> 5;
  const int lane = tid & 31;

  // ---- Phase 0: build twiddle tables, stage fh slice and W_mode into LDS ----
  for (int idx = tid; idx < 4096; idx += 256) {
    const int r = idx >> 6;          // k' / b / j depending on buffer
    const int k = idx & 63;
    const int t = (r * k) & 63;      // angle index mod 64 keeps args small
    float s, c;
    __sincosf(0.09817477042468103870f * (float)t, &s, &c);  // 2*pi/64 * t
    Cs[r * LDSS + k] = c;
    Sn[r * LDSS + k] = -s;

    A[r * LDSS + k] = fh[((size_t)r * NDIM + (size_t)mode) * KDIM + k];

    const v2f wvv = *(const v2f*)(wts + ((size_t)mode * 4096 + (size_t)idx) * 2);
    Wr[r * LDSS + k] = wvv.x;
    Wi[r * LDSS + k] = wvv.y;
  }
  __syncthreads();

  // ---- Phase 1: G = A @ DFT^T   (Gr = A*Cos^T, Gi = A*Sn^T) ----
  for (int tt = 0; tt < 2; ++tt) {
    const int t  = wv + 8 * tt;            // 16 tiles, 2 per wave
    const int rt = (t >> 2) * 16;
    const int ct = (t & 3) * 16;
    v8f gr = {}; v8f gi = {};
#pragma unroll
    for (int k0 = 0; k0 < 64; k0 += 4) {
      const v2f a  = ldfrag(A,  rt, lane, k0);
      const v2f bc = ldfrag(Cs, ct, lane, k0);
      const v2f bs = ldfrag(Sn, ct, lane, k0);
      gr = wmma4(a, bc, gr);
      gi = wmma4(a, bs, gi);
    }
    st_tile(Gr, rt, ct, lane, gr);
    st_tile(Gi, rt, ct, lane, gi);
  }
  __syncthreads();

  // ---- Phase 2: T = G @ W^T (complex):  Tr = Gr*Wr^T - Gi*Wi^T ; Ti = Gr*Wi^T + Gi*Wr^T ----
  for (int tt = 0; tt < 2; ++tt) {
    const int t  = wv + 8 * tt;
    const int rt = (t >> 2) * 16;
    const int ct = (t & 3) * 16;
    v8f tr = {}; v8f ti = {};
#pragma unroll
    for (int k0 = 0; k0 < 64; k0 += 4) {
      const v2f ar = ldfrag(Gr, rt, lane, k0);
      const v2f ai = ldfrag(Gi, rt, lane, k0);
      const v2f br = ldfrag(Wr, ct, lane, k0);
      const v2f bi = ldfrag(Wi, ct, lane, k0);
      v2f nbi; nbi.x = -bi.x; nbi.y = -bi.y;
      tr = wmma4(ar, br, tr);
      ti = wmma4(ar, bi, ti);
      tr = wmma4(ai, nbi, tr);
      ti = wmma4(ai, br, ti);
    }
    st_tile(Tr, rt, ct, lane, tr);
    st_tile(Ti, rt, ct, lane, ti);
  }
  __syncthreads();

  // ---- Phase 3: out = (1/64) * T @ conj(DFT)^T ----
  // ifft basis: exp(+i*th) = Cos - i*Sn  =>  Or = Tr*Cos^T + Ti*Sn^T ; Oi = Ti*Cos^T - Tr*Sn^T
  for (int tt = 0; tt < 2; ++tt) {
    const int t  = wv + 8 * tt;
    const int rt = (t >> 2) * 16;
    const int ct = (t & 3) * 16;
    v8f orr = {}; v8f oii = {};
#pragma unroll
    for (int k0 = 0; k0 < 64; k0 += 4) {
      const v2f ar = ldfrag(Tr, rt, lane, k0);
      const v2f ai = ldfrag(Ti, rt, lane, k0);
      const v2f bc = ldfrag(Cs, ct, lane, k0);
      const v2f bs = ldfrag(Sn, ct, lane, k0);
      v2f nbs; nbs.x = -bs.x; nbs.y = -bs.y;
      orr = wmma4(ar, bc, orr);
      orr = wmma4(ai, bs, orr);
      oii = wmma4(ai, bc, oii);
      oii = wmma4(ar, nbs, oii);
    }
    // Scatter: row = batch b, col = output freq j ; interleaved complex64, scaled 1/64
    const int n  = lane & 15;
    const int hi = lane >> 4;
#pragma unroll
    for (int v = 0; v < 8; ++v) {
      const int b = rt + v + 8 * hi;
      const int j = ct + n;
      const size_t off = (((size_t)b * NMODES + (size_t)mode) * KDIM + (size_t)j) * 2;
      v2f o; o.x = orr[v] * 0.015625f; o.y = oii[v] * 0.015625f;
      *(v2f*)(out + off) = o;
    }
  }
}

extern "C" void kernel_launch(void* const* d_in, const int* in_sizes, int n_in,
                              void* d_out, int out_size, void* d_ws, size_t ws_size,
                              hipStream_t stream) {
  (void)in_sizes; (void)n_in; (void)d_ws; (void)ws_size; (void)out_size;
  const float* fh  = (const float*)d_in[0];   // [64,8192,64] f32
  const float* wts = (const float*)d_in[1];   // [2048,64,64] complex64
  float* out = (float*)d_out;                 // [64,2048,64] complex64
  fno_mode_kernel<<<dim3(NMODES), dim3(256), 0, stream>>>(fh, wts, out);
}